// spatial_gate_attention_22814866277112
// MI455X (gfx1250) — compile-verified
//
#include <hip/hip_runtime.h>

// Problem constants (match reference's setup_inputs()).
#define M_NODES  200000
#define C_FEAT   256
#define E_EDGES  6400000
#define G_GRAPHS 2048

typedef float v2f __attribute__((ext_vector_type(2)));
typedef float v8f __attribute__((ext_vector_type(8)));

// Float atomic max via int-max / uint-min punning (valid with -inf init).
__device__ __forceinline__ void atomic_max_f32(float* addr, float v) {
    if (v >= 0.0f) atomicMax((int*)addr, __float_as_int(v));
    else           atomicMin((unsigned int*)addr, __float_as_uint(v));
}

// K0: zero degree counts, init segment max/sum accumulators.
__global__ void k0_init(unsigned* __restrict__ deg_u,
                        float* __restrict__ gmax_imp, float* __restrict__ gsum_imp,
                        float* __restrict__ gmax_deg, float* __restrict__ gsum_deg) {
    int i = blockIdx.x * blockDim.x + threadIdx.x;
    if (i < M_NODES) deg_u[i] = 0u;
    if (i < G_GRAPHS) {
        gmax_imp[i] = -__builtin_inff();
        gmax_deg[i] = -__builtin_inff();
        gsum_imp[i] = 0.0f;
        gsum_deg[i] = 0.0f;
    }
}

// K1: imp = x @ W^T + b via v_wmma_f32_16x16x4_f32.
// One wave per 16-row tile. A-layout (ISA 7.12.2, 32-bit A 16x4):
//   lanes 0-15: VGPR0=K0, VGPR1=K1 ; lanes 16-31: VGPR0=K2, VGPR1=K3.
// B is W[k] broadcast across all 16 columns, striped with the same K pattern,
// so D[m][n] == dot(x[m,:], W) for every column n. Lanes 0/16 hold N==0.
__global__ void k1_gemv_wmma(const float* __restrict__ x,
                             const float* __restrict__ W,
                             const float* __restrict__ bptr,
                             float* __restrict__ imp) {
    const int lane = threadIdx.x & 31;
    const int wave = threadIdx.x >> 5;
    const int tile = blockIdx.x * (blockDim.x >> 5) + wave;
    const int rowBase = tile * 16;
    if (rowBase >= M_NODES) return;            // wave-uniform exit; EXEC stays full
    const int r  = lane & 15;
    const int h2 = (lane >> 4) << 1;           // 0 for lanes 0-15, 2 for lanes 16-31
    const long rowOff = (long)(rowBase + r) * C_FEAT;

    v8f acc = {};
#pragma unroll 4
    for (int k = 0; k < C_FEAT; k += 4) {
        const int kk = k + h2;
        v2f a  = *(const v2f*)(x + rowOff + kk);   // A[r][kk], A[r][kk+1]
        v2f bb = *(const v2f*)(W + kk);            // B rows kk, kk+1 (bcast over N)
        acc = __builtin_amdgcn_wmma_f32_16x16x4_f32(
            /*neg_a=*/false, a, /*neg_b=*/false, bb,
            /*c_mod=*/(short)0, acc, /*reuse_a=*/false, /*reuse_b=*/false);
    }

    if (lane == 0 || lane == 16) {             // these lanes hold column N==0
        const float bias = bptr[0];
        const int base = rowBase + (lane >> 4) * 8;
#pragma unroll
        for (int i2 = 0; i2 < 8; ++i2) imp[base + i2] = acc[i2] + bias;
    }
}

// K2: out-degree of edge sources (integer scatter-add).
__global__ void k2_degree(const long long* __restrict__ edge_src,
                          unsigned* __restrict__ deg_u) {
    int e = blockIdx.x * blockDim.x + threadIdx.x;
    if (e < E_EDGES) atomicAdd(&deg_u[(int)edge_src[e]], 1u);
}

// K3: convert degree counts to float in place; per-segment maxes of imp & deg.
__global__ void k3_segmax(const float* __restrict__ imp,
                          const unsigned* deg_u, float* deg_f,   // alias same buffer
                          const long long* __restrict__ batch,
                          float* __restrict__ gmax_imp, float* __restrict__ gmax_deg) {
    int i = blockIdx.x * blockDim.x + threadIdx.x;
    if (i >= M_NODES) return;
    const int b = (int)batch[i];
    const float d = (float)deg_u[i];
    deg_f[i] = d;
    atomic_max_f32(&gmax_imp[b], imp[i]);
    atomic_max_f32(&gmax_deg[b], d);
}

// K4: e = exp(v - segmax), overwrite in place, accumulate segment sums.
__global__ void k4_expsum(float* __restrict__ imp, float* __restrict__ deg_f,
                          const long long* __restrict__ batch,
                          const float* __restrict__ gmax_imp,
                          const float* __restrict__ gmax_deg,
                          float* __restrict__ gsum_imp, float* __restrict__ gsum_deg) {
    int i = blockIdx.x * blockDim.x + threadIdx.x;
    if (i >= M_NODES) return;
    const int b = (int)batch[i];
    const float ei = expf(imp[i]   - gmax_imp[b]);
    const float ed = expf(deg_f[i] - gmax_deg[b]);
    imp[i]   = ei;
    deg_f[i] = ed;
    atomicAdd(&gsum_imp[b], ei);
    atomicAdd(&gsum_deg[b], ed);
}

// K5: out[i,:] = (imp_e[i]/gsum_imp[b] + deg_e[i]/gsum_deg[b]) * x[i,:]
// 4 rows per 256-thread block; one float4 (16B) per thread. Dominant pass:
// 205 MB read + 205 MB write of pure streaming.
__global__ void k5_scale(const float* __restrict__ x,
                         const float* __restrict__ imp, const float* __restrict__ deg_f,
                         const long long* __restrict__ batch,
                         const float* __restrict__ gsum_imp, const float* __restrict__ gsum_deg,
                         float* __restrict__ out) {
    const int row = blockIdx.x * 4 + (threadIdx.x >> 6);
    const int c4  = (threadIdx.x & 63) << 2;
    if (row >= M_NODES) return;
    const int b = (int)batch[row];
    const float gate = imp[row] / gsum_imp[b] + deg_f[row] / gsum_deg[b];
    const long off = (long)row * C_FEAT + c4;
    if (row + 4 < M_NODES)
        __builtin_prefetch(x + off + 4 * C_FEAT, 0, 1);   // global_prefetch_b8
    const float4 v = *(const float4*)(x + off);
    float4 o;
    o.x = gate * v.x; o.y = gate * v.y; o.z = gate * v.z; o.w = gate * v.w;
    *(float4*)(out + off) = o;
}

extern "C" void kernel_launch(void* const* d_in, const int* in_sizes, int n_in,
                              void* d_out, int out_size, void* d_ws, size_t ws_size,
                              hipStream_t stream) {
    const float*     x     = (const float*)d_in[0];
    const long long* batch = (const long long*)d_in[1];
    const long long* edge  = (const long long*)d_in[2];   // [2,E]; row 0 = sources
    const float*     W     = (const float*)d_in[3];
    const float*     bias  = (const float*)d_in[4];
    float*           out   = (float*)d_out;

    // Workspace layout: imp[M] | deg[M] (u32 then f32 in place) | 4 x [G]
    float* ws        = (float*)d_ws;
    float* imp       = ws;
    float* deg_f     = ws + M_NODES;
    unsigned* deg_u  = (unsigned*)deg_f;
    float* gmax_imp  = ws + 2 * (size_t)M_NODES;
    float* gsum_imp  = gmax_imp + G_GRAPHS;
    float* gmax_deg  = gsum_imp + G_GRAPHS;
    float* gsum_deg  = gmax_deg + G_GRAPHS;

    const int T = 256;
    const int gM = (M_NODES + T - 1) / T;       // 782
    const int gE = (E_EDGES + T - 1) / T;       // 25000
    const int g1 = (M_NODES + 127) / 128;       // 1563 (8 waves * 16 rows per block)
    const int g5 = (M_NODES + 3) / 4;           // 50000

    k0_init     <<<gM, T, 0, stream>>>(deg_u, gmax_imp, gsum_imp, gmax_deg, gsum_deg);
    k1_gemv_wmma<<<g1, T, 0, stream>>>(x, W, bias, imp);
    k2_degree   <<<gE, T, 0, stream>>>(edge, deg_u);
    k3_segmax   <<<gM, T, 0, stream>>>(imp, deg_u, deg_f, batch, gmax_imp, gmax_deg);
    k4_expsum   <<<gM, T, 0, stream>>>(imp, deg_f, batch, gmax_imp, gmax_deg,
                                       gsum_imp, gsum_deg);
    k5_scale    <<<g5, T, 0, stream>>>(x, imp, deg_f, batch, gsum_imp, gsum_deg, out);
}